// MaxMinQNet_64433099374742
// MI455X (gfx1250) — compile-verified
//
#include <hip/hip_runtime.h>

typedef __attribute__((ext_vector_type(2))) float v2f;
typedef __attribute__((ext_vector_type(8))) float v8f;

#define B_   256
#define OBS_ 256
#define H_   1024
#define A_   18

// ---------------------------------------------------------------------------
// Kernel 1: fc_in  h1[m,n] = sum_k x[m,k] * W_in[n,k] + b_in[n]
// One wave per 32x32 output tile, V_WMMA_F32_16X16X4_F32, exact f32 math.
// A-frag/B-frag per ISA layout: lane(l16,half) holds contiguous float2 at
// row*ld + k + 2*half.
// ---------------------------------------------------------------------------
__global__ __launch_bounds__(32) void k_fc_in(const float* __restrict__ x,
                                              const float* __restrict__ Win,
                                              const float* __restrict__ bin,
                                              float* __restrict__ h1) {
  const int lane = threadIdx.x & 31;
  const int half = lane >> 4;
  const int l16  = lane & 15;
  const int m0   = blockIdx.x * 32;
  const int n0   = blockIdx.y * 32;

  v8f c00 = {}, c01 = {}, c10 = {}, c11 = {};

  const float* a0 = x   + (m0 +      l16) * OBS_ + 2 * half;
  const float* a1 = x   + (m0 + 16 + l16) * OBS_ + 2 * half;
  const float* b0 = Win + (n0 +      l16) * OBS_ + 2 * half;
  const float* b1 = Win + (n0 + 16 + l16) * OBS_ + 2 * half;

  for (int k = 0; k < OBS_; k += 4) {
    v2f af0 = *(const v2f*)(a0 + k);
    v2f af1 = *(const v2f*)(a1 + k);
    v2f bf0 = *(const v2f*)(b0 + k);
    v2f bf1 = *(const v2f*)(b1 + k);
    c00 = __builtin_amdgcn_wmma_f32_16x16x4_f32(false, af0, false, bf0, (short)0, c00, false, false);
    c01 = __builtin_amdgcn_wmma_f32_16x16x4_f32(false, af0, false, bf1, (short)0, c01, false, false);
    c10 = __builtin_amdgcn_wmma_f32_16x16x4_f32(false, af1, false, bf0, (short)0, c10, false, false);
    c11 = __builtin_amdgcn_wmma_f32_16x16x4_f32(false, af1, false, bf1, (short)0, c11, false, false);
  }

  // C/D layout: vgpr v -> row m0 + 16*i + v + 8*half, col n0 + 16*j + l16
  const int col0 = n0 + l16;
  const int col1 = n0 + 16 + l16;
  const float bias0 = bin[col0];
  const float bias1 = bin[col1];
#pragma unroll
  for (int v = 0; v < 8; ++v) {
    const int r0 = m0 + v + 8 * half;
    const int r1 = m0 + 16 + v + 8 * half;
    h1[r0 * H_ + col0] = c00[v] + bias0;
    h1[r0 * H_ + col1] = c01[v] + bias1;
    h1[r1 * H_ + col0] = c10[v] + bias0;
    h1[r1 * H_ + col1] = c11[v] + bias1;
  }
}

// ---------------------------------------------------------------------------
// Kernels 2/3: max-plus / min-plus semiring matmul (NOT expressible as WMMA).
//   out[b,o] = max_i (W[o,i] + h[b,i])   (or min)
// Block: 256 threads (8 wave32), tile = 16 batch x 64 outputs.
// h-tile (16x1024 f32 = 64KB, contiguous) staged once in LDS.
// Each wave owns FOUR output columns at a time so every hs b128 LDS read is
// reused 4x (DS:VALU = 1:32, safely VALU-bound); lanes split the i-dimension
// with coalesced float4 streams of the W rows, then butterfly reduce.
// ---------------------------------------------------------------------------
template <int IS_MAX>
__global__ __launch_bounds__(256) void k_semiring(const float* __restrict__ hin,
                                                  const float* __restrict__ W,
                                                  float* __restrict__ hout) {
  __shared__ float hs[16 * H_];  // 64 KB
  const int b0 = blockIdx.x * 16;
  const int o0 = blockIdx.y * 64;

  // Tile of 16 consecutive h rows is one contiguous 64KB block: flat copy.
  {
    const float4* src = (const float4*)(hin + b0 * H_);
    float4* dst = (float4*)hs;
    for (int idx = threadIdx.x; idx < (16 * H_) / 4; idx += 256) dst[idx] = src[idx];
  }
  __syncthreads();

  const int wave = threadIdx.x >> 5;
  const int lane = threadIdx.x & 31;

  for (int ot = 0; ot < 2; ++ot) {
    const int ob = o0 + wave * 8 + ot * 4;  // 4 consecutive output columns
    const float* wbase = W + (size_t)ob * H_;

    float acc[4][16];
#pragma unroll
    for (int oi = 0; oi < 4; ++oi)
#pragma unroll
      for (int b = 0; b < 16; ++b)
        acc[oi][b] = IS_MAX ? -3.402823466e38f : 3.402823466e38f;

    for (int c = 0; c < H_ / 128; ++c) {
      const int io = c * 128 + lane * 4;
      float4 w[4];
#pragma unroll
      for (int oi = 0; oi < 4; ++oi)
        w[oi] = *(const float4*)(wbase + oi * H_ + io);  // coalesced 512B/wave

#pragma unroll
      for (int b = 0; b < 16; ++b) {
        const float4 hv = *(const float4*)(hs + b * H_ + io);
#pragma unroll
        for (int oi = 0; oi < 4; ++oi) {
          if (IS_MAX) {
            float t0 = fmaxf(w[oi].x + hv.x, w[oi].y + hv.y);
            float t1 = fmaxf(w[oi].z + hv.z, w[oi].w + hv.w);
            acc[oi][b] = fmaxf(acc[oi][b], fmaxf(t0, t1));
          } else {
            float t0 = fminf(w[oi].x + hv.x, w[oi].y + hv.y);
            float t1 = fminf(w[oi].z + hv.z, w[oi].w + hv.w);
            acc[oi][b] = fminf(acc[oi][b], fminf(t0, t1));
          }
        }
      }
    }

    // wave32 butterfly reduction over the i-slices
#pragma unroll
    for (int oi = 0; oi < 4; ++oi) {
#pragma unroll
      for (int b = 0; b < 16; ++b) {
        float v = acc[oi][b];
#pragma unroll
        for (int off = 16; off > 0; off >>= 1) {
          const float other = __shfl_xor(v, off, 32);
          v = IS_MAX ? fmaxf(v, other) : fminf(v, other);
        }
        if (lane == 0) hout[(b0 + b) * H_ + ob + oi] = v;
      }
    }
  }
}

// ---------------------------------------------------------------------------
// Kernel 4: fc_out  q[m,a] = sum_i h[m,i] * W_out[a,i] + b_out[a]
// N=18 padded to two 16-wide WMMA tiles; out-of-range columns clamped to a
// dummy row (uniform select, no EXEC divergence before WMMA) and not stored.
// ---------------------------------------------------------------------------
__global__ __launch_bounds__(32) void k_fc_out(const float* __restrict__ h,
                                               const float* __restrict__ Wout,
                                               const float* __restrict__ bout,
                                               float* __restrict__ q) {
  const int lane = threadIdx.x & 31;
  const int half = lane >> 4;
  const int l16  = lane & 15;
  const int m0   = blockIdx.x * 16;
  const int n0   = blockIdx.y * 16;

  const int col  = n0 + l16;
  const int bcol = (col < A_) ? col : 0;  // v_cndmask, keeps EXEC all-ones

  v8f c = {};
  const float* ap = h    + (m0 + l16) * H_ + 2 * half;
  const float* bp = Wout + bcol * H_       + 2 * half;

  for (int k = 0; k < H_; k += 4) {
    v2f a = *(const v2f*)(ap + k);
    v2f b = *(const v2f*)(bp + k);
    c = __builtin_amdgcn_wmma_f32_16x16x4_f32(false, a, false, b, (short)0, c, false, false);
  }

  if (col < A_) {
    const float bias = bout[col];
#pragma unroll
    for (int v = 0; v < 8; ++v) q[(m0 + v + 8 * half) * A_ + col] = c[v] + bias;
  }
}

// ---------------------------------------------------------------------------
extern "C" void kernel_launch(void* const* d_in, const int* in_sizes, int n_in,
                              void* d_out, int out_size, void* d_ws, size_t ws_size,
                              hipStream_t stream) {
  (void)in_sizes; (void)n_in; (void)out_size; (void)ws_size;
  const float* x    = (const float*)d_in[0];
  const float* Win  = (const float*)d_in[1];
  const float* bin  = (const float*)d_in[2];
  const float* Wmax = (const float*)d_in[3];
  const float* Wmin = (const float*)d_in[4];
  const float* Wout = (const float*)d_in[5];
  const float* bout = (const float*)d_in[6];
  float* q  = (float*)d_out;
  float* h1 = (float*)d_ws;         // B_*H_ f32 = 1 MB
  float* h2 = h1 + (size_t)B_ * H_; // 1 MB

  k_fc_in   <<<dim3(B_ / 32, H_ / 32), 32,  0, stream>>>(x,  Win,  bin,  h1);
  k_semiring<1><<<dim3(B_ / 16, H_ / 64), 256, 0, stream>>>(h1, Wmax, h2);
  k_semiring<0><<<dim3(B_ / 16, H_ / 64), 256, 0, stream>>>(h2, Wmin, h1);
  k_fc_out  <<<dim3(B_ / 16, 2),       32,  0, stream>>>(h1, Wout, bout, q);
}